// QLSTM_65481071399621
// MI455X (gfx1250) — compile-verified
//
#include <hip/hip_runtime.h>

// ---------------------------------------------------------------------------
// QLSTM forward for MI455X (gfx1250, wave32).
//  Per time step (sequential dependence through hx/cx):
//   K1 angles_kernel : ang[64,48] = [x_t ; hx] @ lin_w^T + lin_b   (WMMA f32 16x16x4)
//   K2 qgate_kernel  : 256 WGs (g,b); 4096-amp complex state split
//                      registers (bits 11-8) / lanes (bits 4-0, shfl_xor) /
//                      waves (bits 7-5, LDS butterflies). Only 3 LDS-butterfly
//                      wires per layer; perm gathers into registers.
//   K3 proj_gate_kernel : pr = q @ proj_w^T (WMMA f32 16x16x4, 4 accumulators),
//                      LSTM gating fused on the D-fragment, writes outs/hx/cx
// ---------------------------------------------------------------------------

#define NW      12
#define QDIM    4096        // 1 << NW
#define SEQn    64
#define BATCHn  64
#define INF_    128         // IN_DIM
#define HIDn    128
#define COMBK   256         // IN_DIM + HID
#define NCOL    48          // 4 gates * 12 wires

typedef __attribute__((ext_vector_type(2))) float v2f;
typedef __attribute__((ext_vector_type(8))) float v8f;

#if defined(__HIP_DEVICE_COMPILE__) && __has_builtin(__builtin_amdgcn_wmma_f32_16x16x4_f32)
#define HAVE_WMMA4 1
#else
#define HAVE_WMMA4 0
#endif

__device__ __forceinline__ float sigmoidf_(float x) { return 1.0f / (1.0f + __expf(-x)); }
__device__ __forceinline__ v2f ld2(const float* p) { return *(const v2f*)p; }

// ---------------------------------------------------------------------------
// K0: zero hx / cx (t==0 state); harness poisons d_out, so we must init.
// ---------------------------------------------------------------------------
__global__ void init_state_kernel(float* __restrict__ hx, float* __restrict__ cx) {
  const int i = blockIdx.x * blockDim.x + threadIdx.x;
  if (i < BATCHn * HIDn) { hx[i] = 0.0f; cx[i] = 0.0f; }
}

// ---------------------------------------------------------------------------
// K1: ang[b][col] = sum_f lin_w[col][f] * comb[b][f] + lin_b[col]
//     comb[b][:] = [x_t[b][:], hx[b][:]].  One block, 12 waves; wave (bm,nn)
//     does the 16x16 tile with 64 V_WMMA_F32_16X16X4_F32, branch-free: the
//     K loop is split into the x half and the hx half (hx always valid).
// ---------------------------------------------------------------------------
__global__ __launch_bounds__(384) void angles_kernel(
    const float* __restrict__ xt,      // [64][128]  (x at step t)
    const float* __restrict__ lin_w,   // [48][256]
    const float* __restrict__ lin_b,   // [48]
    const float* __restrict__ hx,      // [64][128]
    float* __restrict__ ang)           // [64][48]
{
#if HAVE_WMMA4
  const int lane  = threadIdx.x & 31;
  const int wv    = threadIdx.x >> 5;     // 0..11
  const int bm    = wv & 3;               // M tile (batch)
  const int nn    = wv >> 2;              // N tile (0..2)
  const int m     = bm * 16 + (lane & 15);
  const int col   = nn * 16 + (lane & 15);
  const int khalf = (lane >> 4) * 2;      // K sub-offset per half-wave

  const float* ax = xt + m * INF_ + khalf;           // A, x half
  const float* ah = hx + m * HIDn + khalf;           // A, hx half
  const float* bw = lin_w + col * COMBK + khalf;     // B (lin_w row = K-major)

  v8f acc = {};
  #pragma unroll 4
  for (int kk = 0; kk < INF_; kk += 4) {
    const v2f a  = ld2(ax + kk);
    const v2f bv = ld2(bw + kk);
    acc = __builtin_amdgcn_wmma_f32_16x16x4_f32(false, a, false, bv,
                                                (short)0, acc, false, false);
  }
  #pragma unroll 4
  for (int kk = 0; kk < HIDn; kk += 4) {
    const v2f a  = ld2(ah + kk);
    const v2f bv = ld2(bw + INF_ + kk);
    acc = __builtin_amdgcn_wmma_f32_16x16x4_f32(false, a, false, bv,
                                                (short)0, acc, false, false);
  }
  const float bias = lin_b[col];
  #pragma unroll
  for (int j = 0; j < 8; ++j) {
    const int brow = bm * 16 + j + 8 * (lane >> 4);
    ang[brow * NCOL + col] = acc[j] + bias;
  }
#else
  for (int idx = threadIdx.x; idx < NCOL * BATCHn; idx += blockDim.x) {
    const int col = idx / BATCHn, b = idx % BATCHn;
    float s = lin_b[col];
    for (int f = 0; f < INF_; ++f) s += lin_w[col * COMBK + f] * xt[b * INF_ + f];
    for (int f = 0; f < HIDn; ++f) s += lin_w[col * COMBK + INF_ + f] * hx[b * HIDn + f];
    ang[b * NCOL + col] = s;
  }
#endif
}

// ---------------------------------------------------------------------------
// K2: quantum circuit, one workgroup (256 thr) per (g,b).
// Amplitude k = (slot<<8) | tid, slot=0..15 in registers, tid=0..255.
//   bit 11..8 (wires 0..3)  -> register-local butterflies
//   bit  7..5 (wires 4..6)  -> LDS butterflies (cross-wave)
//   bit  4..0 (wires 7..11) -> shfl_xor butterflies (cross-lane)
// RX/RY butterflies are side-symmetric under exchange, so each lane only
// updates its own amplitudes from the shuffled partner values.
// ---------------------------------------------------------------------------
__global__ __launch_bounds__(256) void qgate_kernel(
    const float* __restrict__ ang,    // [64][48]
    const float* __restrict__ theta,  // [4][2][12]
    float* __restrict__ q)            // [64][48]
{
  __shared__ float sre[QDIM];
  __shared__ float sim_[QDIM];
  __shared__ float cA[12], sA[12];
  __shared__ float cT[2][12], sT[2][12];
  __shared__ float red[8][12];

  const int g    = blockIdx.x & 3;
  const int b    = blockIdx.x >> 2;
  const int tid  = threadIdx.x;
  const int lane = tid & 31;
  const int wv   = tid >> 5;

  if (tid < 12) {
    const float h = 0.5f * ang[b * NCOL + g * 12 + tid];
    cA[tid] = __cosf(h); sA[tid] = __sinf(h);
  } else if (tid >= 32 && tid < 56) {
    const int d = (tid - 32) / 12, w = (tid - 32) % 12;
    const float h = 0.5f * theta[(g * 2 + d) * 12 + w];
    cT[d][w] = __cosf(h); sT[d][w] = __sinf(h);
  }
  __syncthreads();

  // |0...0> directly in registers
  float r[16], i[16];
  #pragma unroll
  for (int j = 0; j < 16; ++j) { r[j] = 0.0f; i[j] = 0.0f; }
  if (tid == 0) r[0] = 1.0f;

  // ---------------- RX layer (all 12 RX gates commute) ----------------
  // wires 0..3: register-local (bit 11-w of k lives in slot index)
  #pragma unroll
  for (int w = 0; w < 4; ++w) {
    const int lb = 3 - w;                 // bit inside slot index
    const float c = cA[w], sn = sA[w];
    #pragma unroll
    for (int p = 0; p < 8; ++p) {
      const int j0 = ((p >> lb) << (lb + 1)) | (p & ((1 << lb) - 1));
      const int j1 = j0 + (1 << lb);
      const float r0 = r[j0], i0 = i[j0], r1 = r[j1], i1 = i[j1];
      r[j0] = c * r0 + sn * i1;  i[j0] = c * i0 - sn * r1;
      r[j1] = sn * i0 + c * r1;  i[j1] = c * i1 - sn * r0;
    }
  }
  // wires 7..11: cross-lane via shfl_xor (side-symmetric update)
  #pragma unroll
  for (int w = 7; w < 12; ++w) {
    const int xm = 1 << (11 - w);
    const float c = cA[w], sn = sA[w];
    #pragma unroll
    for (int j = 0; j < 16; ++j) {
      const float ro = __shfl_xor(r[j], xm, 32);
      const float io = __shfl_xor(i[j], xm, 32);
      const float nr = c * r[j] + sn * io;
      const float ni = c * i[j] - sn * ro;
      r[j] = nr; i[j] = ni;
    }
  }
  // spill to LDS, then wires 4..6 (cross-wave) as LDS butterflies
  #pragma unroll
  for (int j = 0; j < 16; ++j) { sre[j * 256 + tid] = r[j]; sim_[j * 256 + tid] = i[j]; }
  __syncthreads();
  #pragma unroll
  for (int w = 4; w < 7; ++w) {
    const int ls = 11 - w;                // 7,6,5
    const float c = cA[w], sn = sA[w];
    #pragma unroll
    for (int jj = 0; jj < 8; ++jj) {
      const int p  = tid + jj * 256;
      const int k0 = ((p >> ls) << (ls + 1)) | (p & ((1 << ls) - 1));
      const int k1 = k0 + (1 << ls);
      const float r0 = sre[k0], i0 = sim_[k0], r1 = sre[k1], i1 = sim_[k1];
      sre[k0] = c * r0 + sn * i1;  sim_[k0] = c * i0 - sn * r1;
      sre[k1] = sn * i0 + c * r1;  sim_[k1] = c * i1 - sn * r0;
    }
    __syncthreads();
  }
  // reload own slots (each location has a unique owner: no race)
  #pragma unroll
  for (int j = 0; j < 16; ++j) { r[j] = sre[j * 256 + tid]; i[j] = sim_[j * 256 + tid]; }

  // ---------------- DEPTH x (RY layer + chain permutation) ----------------
  for (int d = 0; d < 2; ++d) {
    // wires 0..3 register-local
    #pragma unroll
    for (int w = 0; w < 4; ++w) {
      const int lb = 3 - w;
      const float c = cT[d][w], sn = sT[d][w];
      #pragma unroll
      for (int p = 0; p < 8; ++p) {
        const int j0 = ((p >> lb) << (lb + 1)) | (p & ((1 << lb) - 1));
        const int j1 = j0 + (1 << lb);
        const float r0 = r[j0], i0 = i[j0], r1 = r[j1], i1 = i[j1];
        r[j0] = c * r0 - sn * r1;  i[j0] = c * i0 - sn * i1;
        r[j1] = sn * r0 + c * r1;  i[j1] = sn * i0 + c * i1;
      }
    }
    // wires 7..11 cross-lane: new = c*mine + (side ? +s : -s)*other
    #pragma unroll
    for (int w = 7; w < 12; ++w) {
      const int xm = 1 << (11 - w);
      const float c = cT[d][w];
      const float sgn = (lane & xm) ? sT[d][w] : -sT[d][w];
      #pragma unroll
      for (int j = 0; j < 16; ++j) {
        const float ro = __shfl_xor(r[j], xm, 32);
        const float io = __shfl_xor(i[j], xm, 32);
        r[j] = c * r[j] + sgn * ro;
        i[j] = c * i[j] + sgn * io;
      }
    }
    // spill, wires 4..6 in LDS
    #pragma unroll
    for (int j = 0; j < 16; ++j) { sre[j * 256 + tid] = r[j]; sim_[j * 256 + tid] = i[j]; }
    __syncthreads();
    #pragma unroll
    for (int w = 4; w < 7; ++w) {
      const int ls = 11 - w;
      const float c = cT[d][w], sn = sT[d][w];
      #pragma unroll
      for (int jj = 0; jj < 8; ++jj) {
        const int p  = tid + jj * 256;
        const int k0 = ((p >> ls) << (ls + 1)) | (p & ((1 << ls) - 1));
        const int k1 = k0 + (1 << ls);
        const float r0 = sre[k0], i0 = sim_[k0], r1 = sre[k1], i1 = sim_[k1];
        sre[k0] = c * r0 - sn * r1;  sim_[k0] = c * i0 - sn * i1;
        sre[k1] = sn * r0 + c * r1;  sim_[k1] = sn * i0 + c * i1;
      }
      __syncthreads();
    }
    // chain permutation: gather straight into registers
    // CHAIN_PERM = src_(0,1) o ... o src_(10,11) o src_(11,0)
    #pragma unroll
    for (int j = 0; j < 16; ++j) {
      int m = (j << 8) | tid;
      m ^= (m & 1) << 11;                     // CNOT(11,0): pc=0, pt=11
      #pragma unroll
      for (int c2 = 10; c2 >= 0; --c2)        // CNOT(c2,c2+1): pc=11-c2, pt=10-c2
        m ^= ((m >> (11 - c2)) & 1) << (10 - c2);
      r[j] = sre[m]; i[j] = sim_[m];
    }
    __syncthreads();   // all gathers done before LDS is overwritten next iter
  }

  // ---------------- Z expectations from registers ----------------
  float z[12];
  #pragma unroll
  for (int w = 0; w < 12; ++w) z[w] = 0.0f;
  #pragma unroll
  for (int j = 0; j < 16; ++j) {
    const int k = (j << 8) | tid;
    const float p = r[j] * r[j] + i[j] * i[j];
    #pragma unroll
    for (int w = 0; w < 12; ++w)
      z[w] += ((k >> (11 - w)) & 1) ? -p : p;
  }
  #pragma unroll
  for (int w = 0; w < 12; ++w)
    for (int off = 16; off > 0; off >>= 1)
      z[w] += __shfl_xor(z[w], off, 32);
  if (lane == 0)
    for (int w = 0; w < 12; ++w) red[wv][w] = z[w];
  __syncthreads();
  if (tid < 12) {
    float s = 0.0f;
    #pragma unroll
    for (int rr = 0; rr < 8; ++rr) s += red[rr][tid];
    q[b * NCOL + g * 12 + tid] = s;
  }
}

// ---------------------------------------------------------------------------
// K3: pr[g] = q[g] (64x12) @ proj_w[g]^T (12x128) + proj_b[g]; fused gating.
// Grid 4 x 256: block = M tile, wave = N tile; 4 WMMA accumulators (f,i,g,o),
// gating applied directly on the D fragment.
// ---------------------------------------------------------------------------
__global__ __launch_bounds__(256) void proj_gate_kernel(
    const float* __restrict__ q,        // [64][48]
    const float* __restrict__ proj_w,   // [4][128][12]
    const float* __restrict__ proj_b,   // [4][128]
    float* __restrict__ out_t,          // [64][128] slice of outs
    float* __restrict__ hx,             // [64][128]
    float* __restrict__ cx)             // [64][128]
{
#if HAVE_WMMA4
  const int lane  = threadIdx.x & 31;
  const int hn    = threadIdx.x >> 5;     // N tile 0..7
  const int bm    = blockIdx.x;           // M tile 0..3
  const int m     = bm * 16 + (lane & 15);
  const int h     = hn * 16 + (lane & 15);
  const int khalf = (lane >> 4) * 2;

  v8f acc[4];
  #pragma unroll
  for (int gg = 0; gg < 4; ++gg) { v8f zz = {}; acc[gg] = zz; }

  #pragma unroll
  for (int gg = 0; gg < 4; ++gg) {
    #pragma unroll
    for (int kt = 0; kt < 3; ++kt) {        // K = 12 = 3 * 4
      const int kb = kt * 4 + khalf;
      const v2f av = ld2(q + m * NCOL + gg * 12 + kb);
      const v2f bv = ld2(proj_w + (gg * 128 + h) * 12 + kb);
      acc[gg] = __builtin_amdgcn_wmma_f32_16x16x4_f32(false, av, false, bv,
                                                      (short)0, acc[gg], false, false);
    }
  }
  #pragma unroll
  for (int j = 0; j < 8; ++j) {
    const int brow = bm * 16 + j + 8 * (lane >> 4);
    const float f  = sigmoidf_(acc[0][j] + proj_b[0 * 128 + h]);
    const float i  = sigmoidf_(acc[1][j] + proj_b[1 * 128 + h]);
    const float gv = tanhf(acc[2][j] + proj_b[2 * 128 + h]);
    const float o  = sigmoidf_(acc[3][j] + proj_b[3 * 128 + h]);
    const float cp = cx[brow * HIDn + h];
    const float c  = f * cp + i * gv;
    const float hh = o * tanhf(c);
    cx[brow * HIDn + h] = c;
    hx[brow * HIDn + h] = hh;
    out_t[brow * HIDn + h] = hh;
  }
#else
  for (int idx = blockIdx.x * blockDim.x + threadIdx.x; idx < BATCHn * HIDn;
       idx += gridDim.x * blockDim.x) {
    const int brow = idx / HIDn, h = idx % HIDn;
    float pr[4];
    for (int gg = 0; gg < 4; ++gg) {
      float s = proj_b[gg * 128 + h];
      for (int w = 0; w < 12; ++w)
        s += proj_w[(gg * 128 + h) * 12 + w] * q[brow * NCOL + gg * 12 + w];
      pr[gg] = s;
    }
    const float f = sigmoidf_(pr[0]), i = sigmoidf_(pr[1]);
    const float gv = tanhf(pr[2]),   o = sigmoidf_(pr[3]);
    const float cp = cx[idx];
    const float c  = f * cp + i * gv;
    const float hh = o * tanhf(c);
    cx[idx] = c; hx[idx] = hh; out_t[idx] = hh;
  }
#endif
}

// ---------------------------------------------------------------------------
extern "C" void kernel_launch(void* const* d_in, const int* in_sizes, int n_in,
                              void* d_out, int out_size, void* d_ws, size_t ws_size,
                              hipStream_t stream) {
  const float* x      = (const float*)d_in[0];   // [64][64][128]
  const float* lin_w  = (const float*)d_in[1];   // [4][12][256]
  const float* lin_b  = (const float*)d_in[2];   // [4][12]
  const float* theta  = (const float*)d_in[3];   // [4][2][12]
  const float* proj_w = (const float*)d_in[4];   // [4][128][12]
  const float* proj_b = (const float*)d_in[5];   // [4][128]

  float* outs = (float*)d_out;                         // [64][64][128]
  float* hx   = outs + (size_t)SEQn * BATCHn * HIDn;   // [64][128]
  float* cx   = hx + (size_t)BATCHn * HIDn;            // [64][128]

  float* ang = (float*)d_ws;                           // [64][48]
  float* q   = ang + BATCHn * NCOL;                    // [64][48]

  init_state_kernel<<<(BATCHn * HIDn + 255) / 256, 256, 0, stream>>>(hx, cx);

  for (int t = 0; t < SEQn; ++t) {
    angles_kernel<<<1, 384, 0, stream>>>(
        x + (size_t)t * BATCHn * INF_, lin_w, lin_b, hx, ang);
    qgate_kernel<<<256, 256, 0, stream>>>(ang, theta, q);
    proj_gate_kernel<<<4, 256, 0, stream>>>(
        q, proj_w, proj_b, outs + (size_t)t * BATCHn * HIDn, hx, cx);
  }
}